// OrthogonalIntegrityAxiom_40570261078450
// MI455X (gfx1250) — compile-verified
//
#include <hip/hip_runtime.h>

typedef __attribute__((ext_vector_type(2))) float v2f;
typedef __attribute__((ext_vector_type(8))) float v8f;

#define EPSV 1e-8f

// ---------------------------------------------------------------------------
// Kernel 1: zero the workspace (accumulators + padded partial arrays + sums).
// ---------------------------------------------------------------------------
__global__ void oia_zero_kernel(float* __restrict__ ws, int n) {
    int i = blockIdx.x * blockDim.x + threadIdx.x;
    int stride = gridDim.x * blockDim.x;
    for (; i < n; i += stride) ws[i] = 0.0f;
}

// ---------------------------------------------------------------------------
// Kernel 2: per-edge unit direction -> 8 even-degree moments + degree,
// scattered to BOTH endpoints with native global_atomic_add_f32 (no return).
// acc layout: node stride 12 floats: {c2, cs, s2, c4, c3s, c2s2, cs3, s4, deg, pad*3}
// ---------------------------------------------------------------------------
__global__ void oia_edge_kernel(const float2* __restrict__ pos,
                                const int* __restrict__ ei,
                                float* __restrict__ acc,
                                int nEdges) {
    int e = blockIdx.x * blockDim.x + threadIdx.x;
    if (e >= nEdges) return;
    // Stream the (linear) edge list ahead of the gathers: global_prefetch_b8.
    __builtin_prefetch(&ei[e + 8192], 0, 1);
    __builtin_prefetch(&ei[nEdges + e + 8192], 0, 1);

    int s = ei[e];
    int t = ei[nEdges + e];
    float2 ps = pos[s];
    float2 pt = pos[t];
    float dx = pt.x - ps.x;
    float dy = pt.y - ps.y;
    float len = sqrtf(dx * dx + dy * dy);
    float inv = 1.0f / fmaxf(len, EPSV);
    float cx = dx * inv;
    float cy = dy * inv;

    float cc = cx * cx, cs = cx * cy, ss = cy * cy;
    float v[9];
    v[0] = cc;      v[1] = cs;      v[2] = ss;
    v[3] = cc * cc; v[4] = cc * cs; v[5] = cc * ss;
    v[6] = cs * ss; v[7] = ss * ss; v[8] = 1.0f;

    float* as_ = acc + (size_t)s * 12;
    float* at_ = acc + (size_t)t * 12;
#pragma unroll
    for (int k = 0; k < 9; ++k) {
        unsafeAtomicAdd(as_ + k, v[k]);   // global_atomic_add_f32 (no return)
        unsafeAtomicAdd(at_ + k, v[k]);
    }
}

// ---------------------------------------------------------------------------
// Kernel 3: per-node closed forms.
//   S2 = a^2 + 2 b^2 + c^2
//   S4 = p^2 + 4 q^2 + 6 r^2 + 4 t^2 + u^2
//   contrib = 0.5*(S2-S4), pairs = 0.5*deg*(deg-1)
// Written into padded arrays (padding stays zero from kernel 1).
// ---------------------------------------------------------------------------
__global__ void oia_node_kernel(const float4* __restrict__ accv,
                                float* __restrict__ contribArr,
                                float* __restrict__ pairsArr,
                                int nNodes) {
    int n = blockIdx.x * blockDim.x + threadIdx.x;
    if (n >= nNodes) return;
    float4 q0 = accv[(size_t)n * 3 + 0];  // c2, cs, s2, c4
    float4 q1 = accv[(size_t)n * 3 + 1];  // c3s, c2s2, cs3, s4
    float4 q2 = accv[(size_t)n * 3 + 2];  // deg, pad...
    float s2 = q0.x * q0.x + 2.0f * q0.y * q0.y + q0.z * q0.z;
    float s4 = q0.w * q0.w + 4.0f * q1.x * q1.x + 6.0f * q1.y * q1.y +
               4.0f * q1.z * q1.z + q1.w * q1.w;
    float deg = q2.x;
    contribArr[n] = 0.5f * (s2 - s4);
    pairsArr[n]   = 0.5f * deg * (deg - 1.0f);
}

// ---------------------------------------------------------------------------
// Kernel 4: fp32 WMMA reduction. With B = ones, V_WMMA_F32_16X16X4_F32 gives
// D[m][n] = rowsum(A[m]) + C[m][n]; accumulating chunks of 64 values into C,
// any single column of C sums to the grand total. Column 0 = s[lane0]+s[lane16].
// Wave-uniform loop => EXEC all-ones at every WMMA (required by ISA).
// ---------------------------------------------------------------------------
__global__ void oia_wmma_reduce_kernel(const float* __restrict__ contribArr,
                                       const float* __restrict__ pairsArr,
                                       float* __restrict__ sums,
                                       int nChunks) {
    const int lane   = threadIdx.x & 31;
    const int wave   = (blockIdx.x * blockDim.x + threadIdx.x) >> 5;
    const int nWaves = (gridDim.x * blockDim.x) >> 5;

    v2f ones; ones.x = 1.0f; ones.y = 1.0f;
    v8f cA = {};
    v8f cB = {};
    for (int chunk = wave; chunk < nChunks; chunk += nWaves) {
        int base = chunk * 64 + lane * 2;               // 8B-aligned
        v2f a = *(const v2f*)(contribArr + base);
        v2f b = *(const v2f*)(pairsArr + base);
        cA = __builtin_amdgcn_wmma_f32_16x16x4_f32(false, a, false, ones,
                                                   (short)0, cA, false, false);
        cB = __builtin_amdgcn_wmma_f32_16x16x4_f32(false, b, false, ones,
                                                   (short)0, cB, false, false);
    }
    float sA = cA[0] + cA[1] + cA[2] + cA[3] + cA[4] + cA[5] + cA[6] + cA[7];
    float sB = cB[0] + cB[1] + cB[2] + cB[3] + cB[4] + cB[5] + cB[6] + cB[7];
    float totA = __shfl(sA, 0, 32) + __shfl(sA, 16, 32);
    float totB = __shfl(sB, 0, 32) + __shfl(sB, 16, 32);
    if (lane == 0) {
        unsafeAtomicAdd(&sums[0], totA);
        unsafeAtomicAdd(&sums[1], totB);
    }
}

// ---------------------------------------------------------------------------
// Kernel 5: loss = pair_loss_sum / max(n_pairs, 1)
// ---------------------------------------------------------------------------
__global__ void oia_finalize_kernel(const float* __restrict__ sums,
                                    float* __restrict__ out) {
    if (blockIdx.x == 0 && threadIdx.x == 0) {
        out[0] = sums[0] / fmaxf(sums[1], 1.0f);
    }
}

// ---------------------------------------------------------------------------
extern "C" void kernel_launch(void* const* d_in, const int* in_sizes, int n_in,
                              void* d_out, int out_size, void* d_ws, size_t ws_size,
                              hipStream_t stream) {
    const float2* pos = (const float2*)d_in[0];  // (1, N, 2) float32
    const int* ei     = (const int*)d_in[2];     // (2, E) int32, row-major
    const int N  = in_sizes[0] / 2;
    const int E  = in_sizes[2] / 2;
    const int NP = ((N + 63) / 64) * 64;         // padded for 64-wide WMMA chunks

    float* ws         = (float*)d_ws;
    float* acc        = ws;                       // N*12 floats
    float* contribArr = ws + (size_t)N * 12;      // NP floats
    float* pairsArr   = contribArr + NP;          // NP floats
    float* sums       = pairsArr + NP;            // 2 floats

    const int totalFloats = N * 12 + 2 * NP + 2;

    oia_zero_kernel<<<512, 256, 0, stream>>>(ws, totalFloats);

    oia_edge_kernel<<<(E + 255) / 256, 256, 0, stream>>>(pos, ei, acc, E);

    oia_node_kernel<<<(N + 255) / 256, 256, 0, stream>>>((const float4*)acc,
                                                         contribArr, pairsArr, N);

    oia_wmma_reduce_kernel<<<16, 256, 0, stream>>>(contribArr, pairsArr, sums,
                                                   NP / 64);

    oia_finalize_kernel<<<1, 32, 0, stream>>>(sums, (float*)d_out);
}